// Mnist_Classifier_85392539779262
// MI455X (gfx1250) — compile-verified
//
#include <hip/hip_runtime.h>
#include <hip/hip_bf16.h>

typedef __attribute__((ext_vector_type(2))) float v2f;
typedef __attribute__((ext_vector_type(8))) float v8f;

// ---------------- problem dims ----------------
// B=32, K_IN=1, K_L1=50, K_L2=100, B_IN=30 (grid60), B_L1=15 (grid30), B_OUT=5 (grid10)
// GEMM per l: M=288 (b,m), K=450 (f,k) padded->452, N=900 (g,n) padded->928

constexpr int AK  = 452;   // padded K
constexpr int NCP = 928;   // padded N (58 tiles of 16)

// ---------------- workspace layout (floats) ----------------
constexpr size_t SZ_DS2 = 15 * 29 * 60;              // 26100
constexpr size_t SZ_DL1 = (size_t)15 * 29 * 29 * 30; // 378450
constexpr size_t SZ_DF2 = 5 * 9 * 9 * 30;            // 12150
constexpr size_t SZ_DI2 = 5 * 9 * 9 * 10;            // 4050
constexpr size_t SZ_XH1 = 32 * 15 * 29 * 2;          // 27840 (complex interleaved)
constexpr size_t SZ_FS2 = (size_t)32 * 50 * 30 * 81 * 2; // 7,776,000
constexpr size_t SZ_A   = (size_t)5 * 288 * AK;      // 650,880 per plane
constexpr size_t SZ_B   = (size_t)5 * NCP * AK;      // 2,097,280 per plane (transposed)
constexpr size_t SZ_C   = (size_t)32 * 100 * 5 * 81; // 1,296,000 per plane

constexpr size_t OFF_DS2 = 0;
constexpr size_t OFF_DL1 = OFF_DS2 + SZ_DS2;
constexpr size_t OFF_DF2 = OFF_DL1 + SZ_DL1;
constexpr size_t OFF_DI2 = OFF_DF2 + SZ_DF2;
constexpr size_t OFF_XH1 = OFF_DI2 + SZ_DI2;
constexpr size_t OFF_FS2 = OFF_XH1 + SZ_XH1;
constexpr size_t OFF_ARE = OFF_FS2 + SZ_FS2;
constexpr size_t OFF_AIM = OFF_ARE + SZ_A;
constexpr size_t OFF_BRE = OFF_AIM + SZ_A;
constexpr size_t OFF_BIM = OFF_BRE + SZ_B;
constexpr size_t OFF_CRE = OFF_BIM + SZ_B;
constexpr size_t OFF_CIM = OFF_CRE + SZ_C;
constexpr size_t OFF_FEAT = OFF_CIM + SZ_C;          // 3200 floats

// ---------------- fp64 Wigner-d helpers (constants are baked numpy in reference) ----
__device__ double factd(int n) {
  double r = 1.0;
  for (int i = 2; i <= n; ++i) r *= (double)i;
  return r;
}

__device__ double wigner_small_d(int l, int mp, int m, double beta) {
  double c = cos(0.5 * beta), s = sin(0.5 * beta);
  double pref = sqrt(factd(l + mp) * factd(l - mp) * factd(l + m) * factd(l - m));
  int t0 = (m - mp) > 0 ? (m - mp) : 0;
  int t1 = (l + m) < (l - mp) ? (l + m) : (l - mp);
  double acc = 0.0;
  for (int t = t0; t <= t1; ++t) {
    double denom = factd(l + m - t) * factd(t) * factd(mp - m + t) * factd(l - mp - t);
    double term = pow(c, (double)(2 * l + m - mp - 2 * t)) *
                  pow(s, (double)(mp - m + 2 * t)) / denom;
    acc += ((mp - m + t) & 1) ? -term : term;
  }
  return pref * acc;
}

__device__ double dh_weight(int bres, double beta) {
  double acc = 0.0;
  for (int j = 0; j < bres; ++j) acc += sin((2.0 * j + 1.0) * beta) / (2.0 * j + 1.0);
  return (2.0 / (double)bres) * sin(beta) * acc;
}

// K0: build all Wigner/quadrature tables (deterministic, each call)
__global__ void build_tables(float* __restrict__ ws) {
  const double PI = 3.14159265358979323846;
  const size_t total = SZ_DS2 + SZ_DL1 + SZ_DF2 + SZ_DI2;
  for (size_t i = (size_t)blockIdx.x * blockDim.x + threadIdx.x; i < total;
       i += (size_t)gridDim.x * blockDim.x) {
    if (i < SZ_DS2) {
      size_t idx = i;
      int k = (int)(idx % 60); idx /= 60;
      int mi = (int)(idx % 29); int l = (int)(idx / 29);
      int mp = mi - 14;
      double beta = PI * (2.0 * k + 1.0) / 120.0;
      double v = 0.0;
      if (mp >= -l && mp <= l)
        v = wigner_small_d(l, mp, 0, beta) * dh_weight(30, beta) * (2.0 * PI / 60.0);
      ws[OFF_DS2 + i] = (float)v;
    } else if (i < SZ_DS2 + SZ_DL1) {
      size_t idx = i - SZ_DS2;
      int k = (int)(idx % 30);
      int ni = (int)((idx / 30) % 29);
      int mi = (int)((idx / 870) % 29);
      int l  = (int)(idx / 25230);
      int mp = mi - 14, m = ni - 14;
      double beta = PI * (2.0 * k + 1.0) / 60.0;
      double v = 0.0;
      if (mp >= -l && mp <= l && m >= -l && m <= l)
        v = wigner_small_d(l, mp, m, beta) * (2.0 * l + 1.0) / (8.0 * PI * PI);
      ws[OFF_DL1 + idx] = (float)v;
    } else if (i < SZ_DS2 + SZ_DL1 + SZ_DF2) {
      size_t idx = i - SZ_DS2 - SZ_DL1;
      int k = (int)(idx % 30);
      int ni = (int)((idx / 30) % 9);
      int mi = (int)((idx / 270) % 9);
      int l  = (int)(idx / 2430);
      int mp = mi - 4, m = ni - 4;
      double beta = PI * (2.0 * k + 1.0) / 60.0;
      double v = 0.0;
      if (mp >= -l && mp <= l && m >= -l && m <= l)
        v = wigner_small_d(l, mp, m, beta) * dh_weight(15, beta) *
            (2.0 * PI / 30.0) * (2.0 * PI / 30.0);
      ws[OFF_DF2 + idx] = (float)v;
    } else {
      size_t idx = i - SZ_DS2 - SZ_DL1 - SZ_DF2;
      int k = (int)(idx % 10);
      int ni = (int)((idx / 10) % 9);
      int mi = (int)((idx / 90) % 9);
      int l  = (int)(idx / 810);
      int mp = mi - 4, m = ni - 4;
      double beta = PI * (2.0 * k + 1.0) / 20.0;
      double v = 0.0;
      if (mp >= -l && mp <= l && m >= -l && m <= l)
        v = wigner_small_d(l, mp, m, beta) * (2.0 * l + 1.0) / (8.0 * PI * PI);
      ws[OFF_DI2 + idx] = (float)v;
    }
  }
}

// K1: xhat1[b,l,m] = sum_k D_S2[l,m,k] * DFT_phi(x)[b,k,mu=m-14]
__global__ void __launch_bounds__(256) s2_project(const float* __restrict__ x,
                                                  float* __restrict__ ws) {
  __shared__ float twc[60], tws[60];
  __shared__ float Fre[60 * 29], Fim[60 * 29];
  const int b = blockIdx.x, tid = threadIdx.x;
  const float TWO_PI = 6.28318530717958647692f;
  for (int i = tid; i < 60; i += 256) {
    float th = TWO_PI * (float)i / 60.f;
    twc[i] = cosf(th); tws[i] = sinf(th);
  }
  __syncthreads();
  const float* xb = x + (size_t)b * 3600;
  for (int i = tid; i < 1740; i += 256) {
    int k = i / 29, mi = i % 29, mu = mi - 14;
    float ar = 0.f, ai = 0.f;
    for (int j = 0; j < 60; ++j) {
      int a = (-mu * j) % 60; if (a < 0) a += 60;   // e^{-2*pi*i*mu*j/60}
      float xv = xb[k * 60 + j];
      ar += xv * twc[a]; ai += xv * tws[a];
    }
    Fre[i] = ar; Fim[i] = ai;
  }
  __syncthreads();
  const float* DS2 = ws + OFF_DS2;
  float* xh = ws + OFF_XH1;
  for (int i = tid; i < 435; i += 256) {
    int l = i / 29, mi = i % 29;
    float ar = 0.f, ai = 0.f;
    for (int k = 0; k < 60; ++k) {
      float d = DS2[(l * 29 + mi) * 60 + k];
      ar += d * Fre[k * 29 + mi];
      ai += d * Fim[k * 29 + mi];
    }
    size_t o = ((size_t)(b * 15 + l) * 29 + mi) * 2;
    xh[o] = ar; xh[o + 1] = ai;
  }
}

// K2: fused S2-conv: yhat -> Wigner-inverse -> iDFT2(30x30) -> ReLU -> selected
// 9x9 forward DFT2. One workgroup per (b,g); yhat LDS-resident (no 173MB h1 in HBM).
constexpr int SM2_FLOATS = 12615 * 2 + 841 * 2 + 870 * 2 + 900 + 270 * 2 + 60; // 30152
constexpr size_t SM2_BYTES = (size_t)SM2_FLOATS * 4;

__global__ void __launch_bounds__(256) s2_conv_fused(const float* __restrict__ psi_re,
                                                     const float* __restrict__ psi_im,
                                                     float* __restrict__ ws) {
  extern __shared__ float sm[];
  float* yre = sm;                 // 12615
  float* yim = yre + 12615;        // 12615
  float* Gre = yim + 12615;        // 841
  float* Gim = Gre + 841;          // 841
  float* Ure = Gim + 841;          // 870
  float* Uim = Ure + 870;          // 870
  float* rbuf = Uim + 870;         // 900
  float* W2re = rbuf + 900;        // 270
  float* W2im = W2re + 270;        // 270
  float* twc = W2im + 270;         // 30
  float* tws = twc + 30;           // 30

  const int bg = blockIdx.x, b = bg / 50, g = bg % 50, tid = threadIdx.x;
  for (int i = tid; i < 30; i += 256) {
    float th = 6.28318530717958647692f * (float)i / 30.f;
    twc[i] = cosf(th); tws[i] = sinf(th);
  }
  const float* xh = ws + OFF_XH1 + (size_t)b * 15 * 29 * 2;
  for (int i = tid; i < 12615; i += 256) {
    int l = i / 841, r = i % 841, mi = r / 29, ni = r % 29;
    int mp = mi - 14, m = ni - 14;
    float vr = 0.f, vi = 0.f;
    if (mp >= -l && mp <= l && m >= -l && m <= l) {
      float xr = xh[(l * 29 + mi) * 2], xi = xh[(l * 29 + mi) * 2 + 1];
      float pr = psi_re[(l * 29 + ni) * 50 + g];   // psi_s2 (15,29,1,50)
      float pi = psi_im[(l * 29 + ni) * 50 + g];
      vr = xr * pr + xi * pi;                      // x * conj(psi)
      vi = xi * pr - xr * pi;
    }
    yre[i] = vr; yim[i] = vi;
  }
  __syncthreads();
  const float* DL1 = ws + OFF_DL1;
  float* FS = ws + OFF_FS2 + (size_t)bg * 30 * 81 * 2;
  for (int kt = 0; kt < 30; ++kt) {
    for (int i = tid; i < 841; i += 256) {   // G[m,n] = sum_l Dinv*yhat
      int mi = i / 29, ni = i % 29;
      float ar = 0.f, ai = 0.f;
      for (int l = 0; l < 15; ++l) {
        float d = DL1[((size_t)(l * 29 + mi) * 29 + ni) * 30 + kt];
        ar += d * yre[l * 841 + i]; ai += d * yim[l * 841 + i];
      }
      Gre[i] = ar; Gim[i] = ai;
    }
    __syncthreads();
    for (int i = tid; i < 870; i += 256) {   // U[p,n] = sum_m G e^{+i mu p}
      int p = i / 29, n = i % 29;
      float ar = 0.f, ai = 0.f;
      for (int mi = 0; mi < 29; ++mi) {
        int a = ((mi - 14) * p) % 30; if (a < 0) a += 30;
        float gr = Gre[mi * 29 + n], gi = Gim[mi * 29 + n];
        ar += gr * twc[a] - gi * tws[a];
        ai += gr * tws[a] + gi * twc[a];
      }
      Ure[i] = ar; Uim[i] = ai;
    }
    __syncthreads();
    for (int i = tid; i < 900; i += 256) {   // r = relu(Re sum_n U e^{+i nu q})
      int p = i / 30, q = i % 30;
      float h = 0.f;
      for (int ni = 0; ni < 29; ++ni) {
        int a = ((ni - 14) * q) % 30; if (a < 0) a += 30;
        h += Ure[p * 29 + ni] * twc[a] - Uim[p * 29 + ni] * tws[a];
      }
      rbuf[i] = h > 0.f ? h : 0.f;
    }
    __syncthreads();
    for (int i = tid; i < 270; i += 256) {   // W2[p,ns] = sum_q r e^{-i nu' q}
      int p = i / 9, ns = i % 9;
      float wr = 0.f, wi = 0.f;
      for (int q = 0; q < 30; ++q) {
        int a = ((ns - 4) * q) % 30; if (a < 0) a += 30;
        float r = rbuf[p * 30 + q];
        wr += r * twc[a]; wi -= r * tws[a];
      }
      W2re[i] = wr; W2im[i] = wi;
    }
    __syncthreads();
    for (int i = tid; i < 81; i += 256) {    // Fsel2[kt,ms,ns] = sum_p W2 e^{-i mu' p}
      int ms = i / 9, ns = i % 9;
      float fr = 0.f, fi = 0.f;
      for (int p = 0; p < 30; ++p) {
        int a = ((ms - 4) * p) % 30; if (a < 0) a += 30;
        float wr = W2re[p * 9 + ns], wi = W2im[p * 9 + ns];
        fr += wr * twc[a] + wi * tws[a];
        fi += wi * twc[a] - wr * tws[a];
      }
      size_t o = ((size_t)kt * 81 + i) * 2;
      FS[o] = fr; FS[o + 1] = fi;
    }
    __syncthreads();
  }
}

// K3: A planes [l][row=(b*9+m)][kk=(f*9+k), padded to 452] re/im; zero-fill pad
__global__ void so3_forward_project(float* __restrict__ ws) {
  const float* DF = ws + OFF_DF2;
  const float* FS = ws + OFF_FS2;
  float* Are = ws + OFF_ARE;
  float* Aim = ws + OFF_AIM;
  const int total = 5 * 288 * AK;  // 650880
  for (int i = blockIdx.x * blockDim.x + threadIdx.x; i < total;
       i += gridDim.x * blockDim.x) {
    int l = i / (288 * AK), rem = i % (288 * AK);
    int row = rem / AK, kk = rem % AK;
    if (kk >= 450) { Are[i] = 0.f; Aim[i] = 0.f; continue; }
    int b = row / 9, m = row % 9, f = kk / 9, k = kk % 9;
    const float* d = DF + ((l * 9 + m) * 9 + k) * 30;
    const float* fs = FS + (((size_t)(b * 50 + f) * 30) * 81 + (size_t)m * 9 + k) * 2;
    float ar = 0.f, ai = 0.f;
    for (int kt = 0; kt < 30; ++kt) {
      float dd = d[kt];
      ar += dd * fs[(size_t)kt * 162];
      ai += dd * fs[(size_t)kt * 162 + 1];
    }
    Are[i] = ar; Aim[i] = ai;
  }
}

// K3b: repack psi_so3 (5,9,9,50,100) -> TRANSPOSED padded B planes
// BT[l][col=(g*9+n), padded 928][kk=(f*9+k), padded 452]; zero-fill pads
__global__ void pack_psi(const float* __restrict__ pre, const float* __restrict__ pim,
                         float* __restrict__ ws) {
  float* Bre = ws + OFF_BRE;
  float* Bim = ws + OFF_BIM;
  const int total = 5 * NCP * AK;  // 2,097,280
  for (int i = blockIdx.x * blockDim.x + threadIdx.x; i < total;
       i += gridDim.x * blockDim.x) {
    int l = i / (NCP * AK), rem = i % (NCP * AK);
    int col = rem / AK, kk = rem % AK;
    if (col >= 900 || kk >= 450) { Bre[i] = 0.f; Bim[i] = 0.f; continue; }
    int f = kk / 9, kp = kk % 9, g = col / 9, n = col % 9;
    size_t s = ((((size_t)l * 9 + n) * 9 + kp) * 50 + f) * 100 + g;
    Bre[i] = pre[s]; Bim[i] = pim[s];
  }
}

// K4: per-l complex GEMM 288x452x928 via V_WMMA_F32_16X16X4_F32
// Each wave: one 16x32 C-tile (two N-tiles, A-frags reused).
// C = A * conj(P):  C_re += a_re*p_re + a_im*p_im ; C_im += a_im*p_re + (-a_re)*p_im
__global__ void __launch_bounds__(128) so3_gemm_wmma(float* __restrict__ ws) {
  const int l = blockIdx.z;                            // 5
  const int mt = blockIdx.y;                           // 18 M-tiles (288/16)
  const int j = blockIdx.x * 4 + (threadIdx.x >> 5);   // N-tile-pair id, need < 29
  if (j >= 29) return;                                 // wave-uniform, before any WMMA
  const int lane = threadIdx.x & 31;
  const int half = lane >> 4;                          // 0: K0/K1, 1: K2/K3
  const int lr = lane & 15;
  const int row = mt * 16 + lr;                        // < 288 always
  const int col0 = j * 32 + lr;                        // tile 2j
  const int col1 = col0 + 16;                          // tile 2j+1

  const float* a_re = ws + OFF_ARE + (size_t)l * (288 * AK) + (size_t)row * AK;
  const float* a_im = ws + OFF_AIM + (size_t)l * (288 * AK) + (size_t)row * AK;
  const float* br0p = ws + OFF_BRE + (size_t)l * (NCP * AK) + (size_t)col0 * AK;
  const float* bi0p = ws + OFF_BIM + (size_t)l * (NCP * AK) + (size_t)col0 * AK;
  const float* br1p = br0p + (size_t)16 * AK;
  const float* bi1p = bi0p + (size_t)16 * AK;

  v8f cr0 = {0.f, 0.f, 0.f, 0.f, 0.f, 0.f, 0.f, 0.f};
  v8f ci0 = {0.f, 0.f, 0.f, 0.f, 0.f, 0.f, 0.f, 0.f};
  v8f cr1 = {0.f, 0.f, 0.f, 0.f, 0.f, 0.f, 0.f, 0.f};
  v8f ci1 = {0.f, 0.f, 0.f, 0.f, 0.f, 0.f, 0.f, 0.f};

  for (int t = 0; t < 113; ++t) {                      // K padded to 452, no guards
    const int k0 = t * 4 + half * 2;                   // 8-byte aligned (k0 even)
    v2f ar = *(const v2f*)(a_re + k0);                 // global_load_b64
    v2f ai = *(const v2f*)(a_im + k0);
    v2f arn; arn.x = -ar.x; arn.y = -ar.y;             // f32 WMMA NEG only allows CNeg
    v2f br0 = *(const v2f*)(br0p + k0);
    v2f bi0 = *(const v2f*)(bi0p + k0);
    v2f br1 = *(const v2f*)(br1p + k0);
    v2f bi1 = *(const v2f*)(bi1p + k0);

    cr0 = __builtin_amdgcn_wmma_f32_16x16x4_f32(false, ar,  false, br0, (short)0, cr0, false, false);
    cr0 = __builtin_amdgcn_wmma_f32_16x16x4_f32(false, ai,  false, bi0, (short)0, cr0, false, false);
    ci0 = __builtin_amdgcn_wmma_f32_16x16x4_f32(false, ai,  false, br0, (short)0, ci0, false, false);
    ci0 = __builtin_amdgcn_wmma_f32_16x16x4_f32(false, arn, false, bi0, (short)0, ci0, false, false);
    cr1 = __builtin_amdgcn_wmma_f32_16x16x4_f32(false, ar,  false, br1, (short)0, cr1, false, false);
    cr1 = __builtin_amdgcn_wmma_f32_16x16x4_f32(false, ai,  false, bi1, (short)0, cr1, false, false);
    ci1 = __builtin_amdgcn_wmma_f32_16x16x4_f32(false, ai,  false, br1, (short)0, ci1, false, false);
    ci1 = __builtin_amdgcn_wmma_f32_16x16x4_f32(false, arn, false, bi1, (short)0, ci1, false, false);
  }

  float* Cre = ws + OFF_CRE;
  float* Cim = ws + OFF_CIM;
  if (col0 < 900) {
    const int g = col0 / 9, n = col0 % 9;
    for (int r = 0; r < 8; ++r) {                      // VGPR r: M=r | M=r+8 by half
      int rowc = mt * 16 + half * 8 + r;
      int bb = rowc / 9, m = rowc % 9;
      size_t o = (((size_t)(bb * 100 + g) * 5 + l) * 9 + m) * 9 + n;
      Cre[o] = cr0[r]; Cim[o] = ci0[r];
    }
  }
  if (col1 < 900) {
    const int g = col1 / 9, n = col1 % 9;
    for (int r = 0; r < 8; ++r) {
      int rowc = mt * 16 + half * 8 + r;
      int bb = rowc / 9, m = rowc % 9;
      size_t o = (((size_t)(bb * 100 + g) * 5 + l) * 9 + m) * 9 + n;
      Cre[o] = cr1[r]; Cim[o] = ci1[r];
    }
  }
}

// K5: per (b,g): Wigner-inverse (l-sum), 10x10 iDFT2, ReLU, quadrature integrate
__global__ void __launch_bounds__(128) so3_inverse_integrate(float* __restrict__ ws) {
  __shared__ float yre[405], yim[405];
  __shared__ float G2re[810], G2im[810];
  __shared__ float twc[10], tws[10];
  __shared__ float red[128];
  const int bg = blockIdx.x, tid = threadIdx.x;
  const float* Cre = ws + OFF_CRE + (size_t)bg * 405;
  const float* Cim = ws + OFF_CIM + (size_t)bg * 405;
  for (int i = tid; i < 405; i += 128) { yre[i] = Cre[i]; yim[i] = Cim[i]; }
  for (int i = tid; i < 10; i += 128) {
    float th = 0.628318530717958647692f * (float)i;
    twc[i] = cosf(th); tws[i] = sinf(th);
  }
  __syncthreads();
  const float* DI = ws + OFF_DI2;
  for (int i = tid; i < 810; i += 128) {
    int kt = i / 81, mn = i % 81;
    float ar = 0.f, ai = 0.f;
    for (int l = 0; l < 5; ++l) {
      float d = DI[(l * 81 + mn) * 10 + kt];
      ar += d * yre[l * 81 + mn]; ai += d * yim[l * 81 + mn];
    }
    G2re[i] = ar; G2im[i] = ai;
  }
  __syncthreads();
  float acc = 0.f;
  const float PI = 3.14159265358979f;
  for (int rowi = tid; rowi < 100; rowi += 128) {
    int kt = rowi / 10, p = rowi % 10;
    float Ur[9], Ui[9];
    for (int n = 0; n < 9; ++n) {
      float ar = 0.f, ai = 0.f;
      for (int mi = 0; mi < 9; ++mi) {
        int a = ((mi - 4) * p) % 10; if (a < 0) a += 10;
        float gr = G2re[kt * 81 + mi * 9 + n], gi = G2im[kt * 81 + mi * 9 + n];
        ar += gr * twc[a] - gi * tws[a];
        ai += gr * tws[a] + gi * twc[a];
      }
      Ur[n] = ar; Ui[n] = ai;
    }
    float s = 0.f;
    for (int q = 0; q < 10; ++q) {
      float h = 0.f;
      for (int n = 0; n < 9; ++n) {
        int a = ((n - 4) * q) % 10; if (a < 0) a += 10;
        h += Ur[n] * twc[a] - Ui[n] * tws[a];
      }
      s += h > 0.f ? h : 0.f;
    }
    float beta = PI * (2.f * kt + 1.f) / 20.f;    // W_INT[kt] inline
    float dsum = 0.f;
    for (int j = 0; j < 5; ++j) dsum += sinf((2.f * j + 1.f) * beta) / (2.f * j + 1.f);
    float wk = 0.4f * sinf(beta) * dsum * (2.f * PI / 10.f) * (2.f * PI / 10.f);
    acc += wk * s;
  }
  red[tid] = acc;
  __syncthreads();
  for (int st = 64; st > 0; st >>= 1) {
    if (tid < st) red[tid] += red[tid + st];
    __syncthreads();
  }
  if (tid == 0) ws[OFF_FEAT + bg] = red[0];
}

// K6: out[b,c] = feats[b,:] . w[c,:] + bias[c]
__global__ void classifier(const float* __restrict__ ws, const float* __restrict__ w,
                           const float* __restrict__ bias, float* __restrict__ out) {
  int i = blockIdx.x * blockDim.x + threadIdx.x;
  if (i >= 320) return;
  int b = i / 10, c = i % 10;
  const float* f = ws + OFF_FEAT + (size_t)b * 100;
  float s = bias[c];
  for (int g = 0; g < 100; ++g) s += f[g] * w[c * 100 + g];
  out[i] = s;
}

extern "C" void kernel_launch(void* const* d_in, const int* in_sizes, int n_in,
                              void* d_out, int out_size, void* d_ws, size_t ws_size,
                              hipStream_t stream) {
  const float* x    = (const float*)d_in[0];
  const float* ps2r = (const float*)d_in[1];
  const float* ps2i = (const float*)d_in[2];
  const float* ps3r = (const float*)d_in[3];
  const float* ps3i = (const float*)d_in[4];
  const float* w    = (const float*)d_in[5];
  const float* bias = (const float*)d_in[6];
  float* ws  = (float*)d_ws;
  float* out = (float*)d_out;

  build_tables<<<512, 256, 0, stream>>>(ws);
  s2_project<<<32, 256, 0, stream>>>(x, ws);
  s2_conv_fused<<<1600, 256, SM2_BYTES, stream>>>(ps2r, ps2i, ws);
  so3_forward_project<<<1024, 256, 0, stream>>>(ws);
  pack_psi<<<2048, 256, 0, stream>>>(ps3r, ps3i, ws);
  so3_gemm_wmma<<<dim3(8, 18, 5), 128, 0, stream>>>(ws);
  so3_inverse_integrate<<<3200, 128, 0, stream>>>(ws);
  classifier<<<2, 256, 0, stream>>>(ws, w, bias, out);
}